// MusicalModel_62560493634090
// MI455X (gfx1250) — compile-verified
//
#include <hip/hip_runtime.h>

typedef float v2f __attribute__((ext_vector_type(2)));
typedef float v8f __attribute__((ext_vector_type(8)));

__device__ __forceinline__ float fast_tanh(float x) {
#if __has_builtin(__builtin_amdgcn_tanhf)
    return __builtin_amdgcn_tanhf(x);        // v_tanh_f32 (gfx1250 TRANS op)
#elif __has_builtin(__builtin_amdgcn_tanh_f32)
    return __builtin_amdgcn_tanh_f32(x);
#else
    return tanhf(x);
#endif
}

// ---------------------------------------------------------------------------
// Workspace layout (floats), computed identically in all kernels:
//   ws[0..15]                  : final hidden states (h_f at 0, h_b at 8)
//   xp region base = ws+16 (uniform base; all accesses use 32-bit indices):
//     f rows 0..len_f          (row len_f = guard row)
//     1 guard row
//     b rows 0..len_b at elem offset 8*(len_f+2)   (row len_b = guard row)
// Total ~ 8*width floats ~ 33.6 MB.
// ---------------------------------------------------------------------------

// Kernel A: fused conv + input-projection for BOTH RNNs via V_WMMA_F32_16X16X4_F32.
//   u[t][k] = piece[t][k] (k<4) | piece[t+1][k-4] (k>=4)
//   xp_f[t][n]           = sum_k Mf[n][k] u[t][k] + cf[n]
//   xp_b[t-index-1][n]   = sum_k Mb[n][k] u[t][k] + cb[n]
// B packs Mf in columns 0..7 and Mb (channel-flipped b_wih) in columns 8..15,
// so one K=4 WMMA pair produces 16 timesteps of inputs for both RNNs.
__global__ __launch_bounds__(32) void xp_precompute_kernel(
    const float* __restrict__ piece,
    const float* __restrict__ conv_w, const float* __restrict__ conv_b,
    const float* __restrict__ f_wih,  const float* __restrict__ f_bih, const float* __restrict__ f_bhh,
    const float* __restrict__ b_wih,  const float* __restrict__ b_bih, const float* __restrict__ b_bhh,
    const int* __restrict__ idx_p,
    float* __restrict__ ws, int L)
{
    const int width = L - 1;
    const int index = *idx_p;
    int len_f = index; if (len_f > width) len_f = width; if (len_f < 0) len_f = 0;
    int len_b = width - index - 1; if (len_b < 0) len_b = 0;

    float* __restrict__ xbase = ws + 16;   // wave-uniform store base (SADDR form)

    const int lane = threadIdx.x;          // 0..31
    const int n    = lane & 15;            // D column / B column
    const int kb   = (lane >> 4) * 2;      // K sub-block for A/B operands

    // ---- per-lane B operands and bias (tiny one-time setup) ----
    float b00 = 0.f, b01 = 0.f, b10 = 0.f, b11 = 0.f, cv = 0.f;
    {
        const int nn = n & 7;
        const int k0 = kb, k1 = kb + 1, k2 = kb + 4, k3 = kb + 5;
        #pragma unroll
        for (int j = 0; j < 8; ++j) {
            // channel flip of the "backward" branch folded into w
            float w = (n < 8) ? f_wih[nn * 8 + j] : b_wih[nn * 8 + (7 - j)];
            b00 += w * conv_w[j * 8 + (k0 & 3) * 2 + (k0 >> 2)];
            b01 += w * conv_w[j * 8 + (k1 & 3) * 2 + (k1 >> 2)];
            b10 += w * conv_w[j * 8 + (k2 & 3) * 2 + (k2 >> 2)];
            b11 += w * conv_w[j * 8 + (k3 & 3) * 2 + (k3 >> 2)];
            cv  += w * conv_b[j];
        }
        cv += (n < 8) ? (f_bih[nn] + f_bhh[nn]) : (b_bih[nn] + b_bhh[nn]);
    }
    v2f B1; B1.x = b00; B1.y = b01;   // K = kb, kb+1
    v2f B2; B2.x = b10; B2.y = b11;   // K = kb+4, kb+5

    // ---- branchless store plumbing: uniform base + per-lane 32-bit index ----
    // row = clamp(t - rowbias, row_lo, row_hi); guard rows swallow out-of-range.
    const int rowbias = (n < 8) ? 0 : (index + 1);
    const int row_lo  = (n < 8) ? 0 : -1;            // -1 -> guard row before b region
    const int row_hi  = (n < 8) ? len_f : len_b;     // guard rows at region ends
    const int chanoff = ((n < 8) ? 0 : 8 * (len_f + 2)) + (n & 7);

    const int m      = lane & 15;                    // A-matrix row
    const int ntiles = (width + 15) >> 4;
    const int rowoff = (lane >> 4) << 3;             // D rows 0..7 vs 8..15

    for (int tile = blockIdx.x; tile < ntiles; tile += gridDim.x) {
        const int t0 = tile << 4;
        int tm = t0 + m; if (tm > L - 2) tm = L - 2;  // clamp loads near the tail
        v2f A1 = *(const v2f*)(piece + (long long)tm * 4 + kb);        // u[t][kb..kb+1]
        v2f A2 = *(const v2f*)(piece + (long long)(tm + 1) * 4 + kb);  // u[t][kb+4..kb+5]

        v8f c;
        #pragma unroll
        for (int r = 0; r < 8; ++r) c[r] = cv;        // bias pre-loaded into C

        c = __builtin_amdgcn_wmma_f32_16x16x4_f32(false, A1, false, B1, (short)0, c, false, false);
        c = __builtin_amdgcn_wmma_f32_16x16x4_f32(false, A2, false, B2, (short)0, c, false, false);

        const int rowbase = t0 + rowoff - rowbias;
        #pragma unroll
        for (int r = 0; r < 8; ++r) {
            int row = rowbase + r;
            row = (row < row_lo) ? row_lo : row;
            row = (row > row_hi) ? row_hi : row;
            xbase[chanoff + row * 8] = c[r];          // 32-bit voffset + uniform saddr
        }
    }
}

// ---------------------------------------------------------------------------
// Kernel B: sequential tanh-RNN scans (critical path).
// block 0 = forward, block 1 = backward (independent waves on separate SIMDs).
// Lanes 0..7 carry h; 8-deep rotating load pipeline hides L2 latency;
// dot split into two FMA chains to shorten the dependent-VALU chain.
// ---------------------------------------------------------------------------
__global__ __launch_bounds__(32) void rnn_scan_kernel(
    float* __restrict__ ws,
    const float* __restrict__ f_whh, const float* __restrict__ b_whh,
    const float* __restrict__ f_bih, const float* __restrict__ f_bhh,
    const float* __restrict__ b_bih, const float* __restrict__ b_bhh,
    const int* __restrict__ idx_p, int L)
{
    const int width = L - 1;
    const int index = *idx_p;
    int len_f = index; if (len_f > width) len_f = width; if (len_f < 0) len_f = 0;
    int len_b = width - index - 1; if (len_b < 0) len_b = 0;

    const bool bwd = (blockIdx.x != 0);
    const float* __restrict__ xb  = ws + 16;                 // uniform base
    const int xoff0               = (bwd ? 8 * (len_f + 2) : 0);
    const int len                 = bwd ? len_b : len_f;
    const float* __restrict__ whh = bwd ? b_whh : f_whh;

    const int i = threadIdx.x & 7;      // lanes 8..31 mirror lanes 0..7
    const int xoff = xoff0 + i;         // per-lane 32-bit element offset
    float w[8];
    #pragma unroll
    for (int j = 0; j < 8; ++j) w[j] = whh[i * 8 + j];

    float h = 0.f;
    if (len <= 0) {
        // reference pads with one all-zero input row: xp_row = b_ih + b_hh
        const float bias = bwd ? (b_bih[i] + b_bhh[i]) : (f_bih[i] + f_bhh[i]);
        h = fast_tanh(bias);
    } else {
        // 8-deep software pipeline of xp-row loads
        float buf[8];
        #pragma unroll
        for (int k = 0; k < 8; ++k) {
            int tt = (k < len) ? k : (len - 1);
            buf[k] = xb[xoff + tt * 8];
        }
        const int full = len & ~7;
        int t = 0;
        for (; t < full; t += 8) {
            #pragma unroll
            for (int k = 0; k < 8; ++k) {
                #define HJ(j) __int_as_float(__builtin_amdgcn_readlane(__float_as_int(h), (j)))
                float a0 = buf[k] + w[0] * HJ(0);
                float a1 = w[1] * HJ(1);
                a0 += w[2] * HJ(2);  a1 += w[3] * HJ(3);
                a0 += w[4] * HJ(4);  a1 += w[5] * HJ(5);
                a0 += w[6] * HJ(6);  a1 += w[7] * HJ(7);
                #undef HJ
                h = fast_tanh(a0 + a1);
                int tp = t + 8 + k; if (tp >= len) tp = len - 1;  // clamped prefetch
                buf[k] = xb[xoff + tp * 8];
            }
        }
        for (int k = 0; k < len - full; ++k) {
            #define HJ(j) __int_as_float(__builtin_amdgcn_readlane(__float_as_int(h), (j)))
            float a0 = buf[k] + w[0] * HJ(0);
            float a1 = w[1] * HJ(1);
            a0 += w[2] * HJ(2);  a1 += w[3] * HJ(3);
            a0 += w[4] * HJ(4);  a1 += w[5] * HJ(5);
            a0 += w[6] * HJ(6);  a1 += w[7] * HJ(7);
            #undef HJ
            h = fast_tanh(a0 + a1);
        }
    }
    if (threadIdx.x < 8) ws[(bwd ? 8 : 0) + threadIdx.x] = h;
}

// ---------------------------------------------------------------------------
// Kernel C: out = fc_w @ outer(h_f, h_b).reshape(64) + fc_b   (4 outputs)
// ---------------------------------------------------------------------------
__global__ __launch_bounds__(32) void head_kernel(
    const float* __restrict__ ws, const float* __restrict__ fc_w,
    const float* __restrict__ fc_b, float* __restrict__ out)
{
    const int o = threadIdx.x;
    if (o < 4) {
        float acc = fc_b[o];
        for (int i = 0; i < 8; ++i) {
            const float f = ws[i];
            #pragma unroll
            for (int j = 0; j < 8; ++j)
                acc += fc_w[o * 64 + i * 8 + j] * f * ws[8 + j];
        }
        out[o] = acc;
    }
}

extern "C" void kernel_launch(void* const* d_in, const int* in_sizes, int n_in,
                              void* d_out, int out_size, void* d_ws, size_t ws_size,
                              hipStream_t stream) {
    const float* piece  = (const float*)d_in[0];
    const float* conv_w = (const float*)d_in[1];
    const float* conv_b = (const float*)d_in[2];
    const float* f_wih  = (const float*)d_in[3];
    const float* f_whh  = (const float*)d_in[4];
    const float* f_bih  = (const float*)d_in[5];
    const float* f_bhh  = (const float*)d_in[6];
    const float* b_wih  = (const float*)d_in[7];
    const float* b_whh  = (const float*)d_in[8];
    const float* b_bih  = (const float*)d_in[9];
    const float* b_bhh  = (const float*)d_in[10];
    const float* fc_w   = (const float*)d_in[11];
    const float* fc_b   = (const float*)d_in[12];
    const int*   idx    = (const int*)d_in[13];
    float* ws  = (float*)d_ws;
    float* out = (float*)d_out;

    const int L = in_sizes[0] / 4;

    xp_precompute_kernel<<<2048, 32, 0, stream>>>(
        piece, conv_w, conv_b, f_wih, f_bih, f_bhh, b_wih, b_bih, b_bhh, idx, ws, L);
    rnn_scan_kernel<<<2, 32, 0, stream>>>(
        ws, f_whh, b_whh, f_bih, f_bhh, b_bih, b_bhh, idx, L);
    head_kernel<<<1, 32, 0, stream>>>(ws, fc_w, fc_b, out);
}